// Group_3762391351882
// MI455X (gfx1250) — compile-verified
//
#include <hip/hip_runtime.h>
#include <hip/hip_bf16.h>
#include <math.h>

typedef __attribute__((ext_vector_type(2))) float v2f;
typedef __attribute__((ext_vector_type(8))) float v8f;

#define BATCH 64
#define NPTS  16384
#define NGRP  128
#define MGRP  32
#define GT    16      // centers per KNN block
#define CHUNK 1024    // columns per chunk staged in LDS

static __device__ __forceinline__ float fInf() { return __builtin_huge_valf(); }

// ---------------------------------------------------------------------------
// Kernel 1: FPS (exact, sequential-in-K, parallel-in-N) + pack (x,y,z,|x|^2)
// One block per batch, 1024 threads, 16 points per thread held in registers.
// ---------------------------------------------------------------------------
__global__ __launch_bounds__(1024) void fps_pack_kernel(
    const float* __restrict__ xyz,      // (B,N,3)
    float* __restrict__ out_center,     // (B,G,3)
    float4* __restrict__ pts4,          // ws: (B,N) packed (x,y,z,|x|^2)
    float4* __restrict__ c4)            // ws: (B,G) packed centers
{
  const int b    = blockIdx.x;
  const int t    = threadIdx.x;
  const int lane = t & 31;
  const int warp = t >> 5;

  __shared__ float redv[32];
  __shared__ int   redi[32];
  __shared__ float bc[3];

  const float* xb = xyz + (size_t)b * NPTS * 3;

  float px[16], py[16], pz[16], mind[16];
#pragma unroll
  for (int j = 0; j < 16; ++j) {
    const int n = t + j * 1024;
    const float x = xb[n * 3 + 0];
    const float y = xb[n * 3 + 1];
    const float z = xb[n * 3 + 2];
    px[j] = x; py[j] = y; pz[j] = z;
    pts4[(size_t)b * NPTS + n] = make_float4(x, y, z, x * x + y * y + z * z);
    mind[j] = fInf();
  }

  // center 0 = point 0 (reference: random_start_point=False)
  float ccx = xb[0], ccy = xb[1], ccz = xb[2];
  if (t == 0) {
    out_center[((size_t)b * NGRP + 0) * 3 + 0] = ccx;
    out_center[((size_t)b * NGRP + 0) * 3 + 1] = ccy;
    out_center[((size_t)b * NGRP + 0) * 3 + 2] = ccz;
    c4[(size_t)b * NGRP + 0] = make_float4(ccx, ccy, ccz, 0.0f);
  }

  for (int k = 1; k < NGRP; ++k) {
    float bv = -fInf();
    int   bi = 0;
#pragma unroll
    for (int j = 0; j < 16; ++j) {
      const float dx = px[j] - ccx, dy = py[j] - ccy, dz = pz[j] - ccz;
      const float d  = dx * dx + dy * dy + dz * dz;
      mind[j] = fminf(mind[j], d);
      if (mind[j] > bv) { bv = mind[j]; bi = t + j * 1024; }  // j asc -> lowest idx on tie
    }
    // wave32 argmax, tie -> lowest index (matches jnp.argmax)
#pragma unroll
    for (int off = 16; off >= 1; off >>= 1) {
      const float ov = __shfl_xor(bv, off, 32);
      const int   oi = __shfl_xor(bi, off, 32);
      if (ov > bv || (ov == bv && oi < bi)) { bv = ov; bi = oi; }
    }
    if (lane == 0) { redv[warp] = bv; redi[warp] = bi; }
    __syncthreads();
    if (warp == 0) {
      bv = redv[lane]; bi = redi[lane];
#pragma unroll
      for (int off = 16; off >= 1; off >>= 1) {
        const float ov = __shfl_xor(bv, off, 32);
        const int   oi = __shfl_xor(bi, off, 32);
        if (ov > bv || (ov == bv && oi < bi)) { bv = ov; bi = oi; }
      }
      if (lane == 0) {
        const int sel = bi;
        const float sx = xb[sel * 3 + 0];
        const float sy = xb[sel * 3 + 1];
        const float sz = xb[sel * 3 + 2];
        bc[0] = sx; bc[1] = sy; bc[2] = sz;
        out_center[((size_t)b * NGRP + k) * 3 + 0] = sx;
        out_center[((size_t)b * NGRP + k) * 3 + 1] = sy;
        out_center[((size_t)b * NGRP + k) * 3 + 2] = sz;
        c4[(size_t)b * NGRP + k] = make_float4(sx, sy, sz, 0.0f);
      }
    }
    __syncthreads();
    ccx = bc[0]; ccy = bc[1]; ccz = bc[2];
  }
}

// ---------------------------------------------------------------------------
// Kernel 2: KNN top-32 via V_WMMA_F32_16X16X4_F32 scores + streaming select,
// then gather. Grid: (G/GT, B). 512 threads = 16 waves; wave r owns center r.
// Score = |x|^2 - 2 c.x  (the per-row |c|^2 constant cannot change top-k).
//   A(16x4): row m = (-2cx, -2cy, -2cz, 1)   B(4x16): col n = (x, y, z, |x|^2)
// B-fragment trick: lanes 0-15 need (K0,K1)=(x,y) = bytes [0,8) of the packed
// float4; lanes 16-31 need (K2,K3)=(z,|x|^2) = bytes [8,16). So each lane does
// ONE b64 load at base + (hi?8:0) -- no duplicate loads, no cndmask shuffling.
// ---------------------------------------------------------------------------
__global__ __launch_bounds__(512) void knn_gather_kernel(
    const float4* __restrict__ pts4,   // (B,N) packed
    const float4* __restrict__ c4,     // (B,G) packed centers
    float* __restrict__ out_nb,        // (B,G,M,3)
    float* __restrict__ out_idx)       // (B,G,M) stored as float
{
  const int b    = blockIdx.y;
  const int g0   = blockIdx.x * GT;
  const int t    = threadIdx.x;
  const int wave = t >> 5;
  const int lane = t & 31;
  const int l15  = lane & 15;
  const bool hi  = lane >= 16;

  extern __shared__ char smem[];
  float* sval = (float*)smem;                        // GT*CHUNK floats (64 KB)
  int*   sidx = (int*)(smem + GT * CHUNK * 4);       // GT*CHUNK ints  (64 KB)

  // A fragment per ISA 16x4 f32 layout:
  //   lanes 0-15: (K0,K1)=(-2cx,-2cy)   lanes 16-31: (K2,K3)=(-2cz, 1)
  const float4 cc = c4[(size_t)b * NGRP + g0 + l15];
  v2f afrag;
  afrag.x = hi ? (-2.0f * cc.z) : (-2.0f * cc.x);
  afrag.y = hi ? 1.0f           : (-2.0f * cc.y);

  // per-lane register-resident sorted (ascending) top-32 candidate list
  float lv[32];
  int   li[32];
#pragma unroll
  for (int j = 0; j < 32; ++j) { lv[j] = fInf(); li[j] = 0; }

  const float4* pb  = pts4 + (size_t)b * NPTS;
  const float2* pb2 = (const float2*)pb;             // 2 float2 per point

  for (int ch = 0; ch < NPTS / CHUNK; ++ch) {
    const int colbase = ch * CHUNK;

    // ---- compute phase: this wave covers 64 columns of the 16-row tile.
    //      Issue all 4 B-fragment loads first (clause-able, progressive
    //      loadcnt waits), then run the 4 WMMAs + LDS stores. ----
    float2 p2[4];
#pragma unroll
    for (int it = 0; it < 4; ++it) {
      const int colloc = wave * 64 + it * 16 + l15;
      p2[it] = pb2[(size_t)(colbase + colloc) * 2 + (hi ? 1 : 0)];
    }
#pragma unroll
    for (int it = 0; it < 4; ++it) {
      const int colloc = wave * 64 + it * 16 + l15;
      v2f bfrag;
      bfrag.x = p2[it].x;
      bfrag.y = p2[it].y;
      v8f acc = {};
      acc = __builtin_amdgcn_wmma_f32_16x16x4_f32(
          /*neg_a=*/false, afrag, /*neg_b=*/false, bfrag,
          /*c_mod=*/(short)0, acc, /*reuse_a=*/false, /*reuse_b=*/false);
      // C layout: VGPR v -> row v (lanes 0-15) / row v+8 (lanes 16-31)
      const int rb = hi ? 8 : 0;
#pragma unroll
      for (int v = 0; v < 8; ++v)
        sval[(rb + v) * CHUNK + colloc] = acc[v];
    }
    if (ch + 1 < NPTS / CHUNK)
      __builtin_prefetch((const void*)(pb + colbase + CHUNK + wave * 64 + lane), 0, 3);
    __syncthreads();

    // ---- select phase: wave owns row `wave`; lane takes blocked cols
    //      [tt*128 + lane*4, +4) so each pass is ONE ds_load_b128 ----
    const float4* srow = (const float4*)&sval[wave * CHUNK];
    for (int tt = 0; tt < CHUNK / 128; ++tt) {
      const float4 s4 = srow[tt * 32 + lane];
      const int    n0 = colbase + tt * 128 + lane * 4;
      const float  sq[4] = {s4.x, s4.y, s4.z, s4.w};
#pragma unroll
      for (int q = 0; q < 4; ++q) {
        const float s = sq[q];
        if (s < lv[31]) {   // beats current per-lane worst -> sorted insert
          float cv = s; int ci = n0 + q;
#pragma unroll
          for (int j = 0; j < 32; ++j) {
            const bool sw = cv < lv[j];
            const float tv = sw ? lv[j] : cv;
            const int   ti = sw ? li[j] : ci;
            lv[j] = sw ? cv : lv[j];
            li[j] = sw ? ci : li[j];
            cv = tv; ci = ti;
          }
        }
      }
    }
    __syncthreads();
  }

  // ---- dump per-lane sorted lists, then 32-way merge (exact top-32) ----
#pragma unroll
  for (int j = 0; j < 32; ++j) {
    sval[wave * CHUNK + lane * 32 + j] = lv[j];
    sidx[wave * CHUNK + lane * 32 + j] = li[j];
  }
  __syncthreads();

  int ptr = 0;
  const int g = g0 + wave;
  for (int k = 0; k < MGRP; ++k) {
    float hv = sval[wave * CHUNK + lane * 32 + ptr];
    int   hn = sidx[wave * CHUNK + lane * 32 + ptr];
    int   wl = lane;
#pragma unroll
    for (int off = 16; off >= 1; off >>= 1) {
      const float ov = __shfl_xor(hv, off, 32);
      const int   on = __shfl_xor(hn, off, 32);
      const int   ow = __shfl_xor(wl, off, 32);
      if (ov < hv || (ov == hv && on < hn)) { hv = ov; hn = on; wl = ow; }
    }
    if (lane == wl) {       // winner pops its head and writes outputs
      ptr++;
      const size_t o = ((size_t)b * NGRP + g) * MGRP + k;
      out_idx[o] = (float)hn;
      const float4 p = pb[hn];
      out_nb[o * 3 + 0] = p.x;
      out_nb[o * 3 + 1] = p.y;
      out_nb[o * 3 + 2] = p.z;
    }
  }
}

// ---------------------------------------------------------------------------
extern "C" void kernel_launch(void* const* d_in, const int* in_sizes, int n_in,
                              void* d_out, int out_size, void* d_ws, size_t ws_size,
                              hipStream_t stream) {
  const float* xyz = (const float*)d_in[0];

  float* out        = (float*)d_out;
  float* out_nb     = out;                                          // B*G*M*3
  float* out_center = out + (size_t)BATCH * NGRP * MGRP * 3;        // B*G*3
  float* out_idx    = out_center + (size_t)BATCH * NGRP * 3;        // B*G*M

  float4* pts4 = (float4*)d_ws;                    // B*N float4  (16 MB)
  float4* c4   = pts4 + (size_t)BATCH * NPTS;      // B*G float4

  fps_pack_kernel<<<BATCH, 1024, 0, stream>>>(xyz, out_center, pts4, c4);

  dim3 grid2(NGRP / GT, BATCH);
  const size_t lds2 = (size_t)GT * CHUNK * 4 * 2;  // 128 KB (scores + idx)
  knn_gather_kernel<<<grid2, 512, lds2, stream>>>(pts4, c4, out_nb, out_idx);
}